// ScaledDotProductAttention_84361747628297
// MI455X (gfx1250) — compile-verified
//
#include <hip/hip_runtime.h>
#include <hip/hip_bf16.h>

// Problem constants (fixed by the reference).
#define BB 64
#define SS 1024
#define DD 64

typedef float v2f __attribute__((ext_vector_type(2)));
typedef float v8f __attribute__((ext_vector_type(8)));
typedef int   v4i __attribute__((vector_size(16)));   // b128 payload type

// Async global->LDS path (gfx1250). Guarded so the file compiles on any
// toolchain; fallback is a plain vector copy through VGPRs.
#if defined(__has_builtin)
#if __has_builtin(__builtin_amdgcn_global_load_async_to_lds_b128)
#define K_HAVE_ASYNC 1
#endif
#endif

#if defined(K_HAVE_ASYNC)
typedef __attribute__((address_space(1))) v4i* g_v4i_p;   // global b128 ptr
typedef __attribute__((address_space(3))) v4i* l_v4i_p;   // LDS b128 ptr
#endif

static __device__ __forceinline__ void async_wait0() {
#if defined(__has_builtin) && __has_builtin(__builtin_amdgcn_s_wait_asynccnt)
    __builtin_amdgcn_s_wait_asynccnt(0);
#else
    asm volatile("s_wait_asynccnt 0x0" ::: "memory");
#endif
}

// D = A(16x4) * B(4x16) + C, f32 exact math on the CDNA5 matrix pipe.
static __device__ __forceinline__ v8f wmma4(v2f a, v2f b, v8f c) {
    return __builtin_amdgcn_wmma_f32_16x16x4_f32(
        /*neg_a=*/false, a, /*neg_b=*/false, b,
        /*c_mod=*/(short)0, c, /*reuse_a=*/false, /*reuse_b=*/false);
}

// ---------------------------------------------------------------------------
// Kernel 1: relative-key bias.  For fixed query position qpos:
//   bias[b, kcol] = sum_d Q[b,qpos,d] * R_k[qpos,kcol,d]
// GEMM over batches: M=64(b), N=128 (k-chunk), K=64(d).  R_k read ONCE total.
// The 128x64 R_k chunk (32 KB) is staged into LDS with coalesced
// global_load_async_to_lds_b128 (16 B per lane), then fragment gathers are
// served from LDS (row stride 68 dwords: 16B-aligned rows, conflict-free
// 16-lane gathers on banks {d0, d0+4, ...}).
// Grid: (8 k-chunks, 1024 qpos), 256 threads = 8 waves; wave w owns n-tile w.
// Result written into the attn region of d_out (used as scores scratch).
// ---------------------------------------------------------------------------
__global__ __launch_bounds__(256) void k_bias_scores(
    const float* __restrict__ q, const float* __restrict__ Rk,
    float* __restrict__ attn)
{
    constexpr int LSTRIDE = DD + 4;           // 68 dwords: 272 B = 17*16 B
    __shared__ float sB[128 * LSTRIDE];       // ~34 KB

    const int kc   = blockIdx.x;              // 0..7  (128-wide k chunk)
    const int qpos = blockIdx.y;              // 0..1023
    const int tid  = threadIdx.x;
    const int wave = tid >> 5;
    const int lane = tid & 31;
    const int g    = lane >> 4;               // half-wave group
    const int ln   = lane & 15;
    const int col  = kc * 128 + wave * 16 + ln;   // this lane's N column

    // ---- Stage R_k[qpos, kc*128 .. +127, :] into LDS -----------------------
    const float* chunk = Rk + ((size_t)qpos * SS + (size_t)kc * 128) * DD;
#pragma unroll
    for (int i = 0; i < 8; ++i) {
        const int idx = i * 256 + tid;        // 2048 x 16B transfers
        const int row = idx >> 4;
        const int seg = idx & 15;
        const float* gp = chunk + row * DD + seg * 4;
        float*       lp = &sB[row * LSTRIDE + seg * 4];
#if defined(K_HAVE_ASYNC)
        __builtin_amdgcn_global_load_async_to_lds_b128(
            (g_v4i_p)(unsigned long long)(const void*)gp,
            (l_v4i_p)(unsigned)(unsigned long long)(void*)lp,
            /*offset=*/0, /*cpol=*/0);
#else
        *(float4*)lp = *(const float4*)gp;
#endif
    }
#if defined(K_HAVE_ASYNC)
    async_wait0();
#endif
    __syncthreads();

    // ---- GEMM --------------------------------------------------------------
    v8f acc[4];
#pragma unroll
    for (int mt = 0; mt < 4; ++mt) acc[mt] = {};

    const int lcol = wave * 16 + ln;          // column within the LDS chunk
#pragma unroll
    for (int ks = 0; ks < 16; ++ks) {         // K = 64 in steps of 4
        const int d0 = 4 * ks + 2 * g;
        v2f bf = *(const v2f*)&sB[lcol * LSTRIDE + d0];
#pragma unroll
        for (int mt = 0; mt < 4; ++mt) {      // 4 batch tiles of 16
            const int b = mt * 16 + ln;
            v2f af = *(const v2f*)(q + ((size_t)b * SS + qpos) * DD + d0);
            acc[mt] = wmma4(af, bf, acc[mt]);
        }
    }

#pragma unroll
    for (int mt = 0; mt < 4; ++mt)
#pragma unroll
        for (int vv = 0; vv < 8; ++vv) {
            const int b = mt * 16 + vv + 8 * g;
            attn[(size_t)b * SS * SS + (size_t)qpos * SS + col] = acc[mt][vv];
        }
}

// ---------------------------------------------------------------------------
// Kernel 2: qk GEMM + bias add + scale + causal mask + row softmax.
// Grid: (64 q-tiles, 64 batches), 128 threads = 4 waves.
// Each wave owns 16 q-rows x 256 k-cols (16 WMMA tiles, 128 acc VGPRs);
// row softmax: 16-lane shfl_xor reduction + LDS cross-wave combine.
// Reads bias from attn buffer (kernel 1), overwrites with final attn.
// ---------------------------------------------------------------------------
__global__ __launch_bounds__(128) void k_qk_softmax(
    const float* __restrict__ q, const float* __restrict__ k,
    const int* __restrict__ od_ptr, float* __restrict__ attn)
{
    const int q0   = blockIdx.x * 16;
    const int b    = blockIdx.y;
    const int tid  = threadIdx.x;
    const int wave = tid >> 5;
    const int lane = tid & 31;
    const int g    = lane >> 4;
    const int ln   = lane & 15;
    const int od   = *od_ptr;               // one_direction flag

    __shared__ float red[4][16];

    // Preload all A fragments: Q[b, q0+ln, :] (16 K-steps x float2).
    v2f afr[16];
#pragma unroll
    for (int ks = 0; ks < 16; ++ks)
        afr[ks] = *(const v2f*)(q + ((size_t)b * SS + q0 + ln) * DD + 4 * ks + 2 * g);

    v8f acc[16];
#pragma unroll
    for (int nt = 0; nt < 16; ++nt) acc[nt] = {};

#pragma unroll
    for (int nt = 0; nt < 16; ++nt) {
        const int cb = wave * 256 + nt * 16;
#pragma unroll
        for (int ks = 0; ks < 16; ++ks) {
            v2f bf = *(const v2f*)(k + ((size_t)b * SS + cb + ln) * DD + 4 * ks + 2 * g);
            acc[nt] = wmma4(afr[ks], bf, acc[nt]);
        }
    }

    // Add relative-key bias, scale by 1/sqrt(64), apply causal mask.
    const float* biasb = attn + (size_t)b * SS * SS;
#pragma unroll
    for (int nt = 0; nt < 16; ++nt) {
        const int cb = wave * 256 + nt * 16 + ln;
#pragma unroll
        for (int vv = 0; vv < 8; ++vv) {
            const int row = q0 + vv + 8 * g;
            float s = (acc[nt][vv] + biasb[(size_t)row * SS + cb]) * 0.125f;
            if (od && (cb > row)) s = -1e9f;
            acc[nt][vv] = s;
        }
    }

    // Row max: local over 16 tiles, then across the 16 lanes of this group,
    // then across the 4 waves via LDS.
    float rmax[8];
#pragma unroll
    for (int vv = 0; vv < 8; ++vv) {
        float m = acc[0][vv];
#pragma unroll
        for (int nt = 1; nt < 16; ++nt) m = fmaxf(m, acc[nt][vv]);
#pragma unroll
        for (int x = 1; x < 16; x <<= 1) m = fmaxf(m, __shfl_xor(m, x, 32));
        if (ln == 0) red[wave][vv + 8 * g] = m;
    }
    __syncthreads();
#pragma unroll
    for (int vv = 0; vv < 8; ++vv) {
        float m = red[0][vv + 8 * g];
#pragma unroll
        for (int w = 1; w < 4; ++w) m = fmaxf(m, red[w][vv + 8 * g]);
        rmax[vv] = m;
    }
    __syncthreads();

    // exp + row sum (same reduction pattern).
    float rinv[8];
#pragma unroll
    for (int vv = 0; vv < 8; ++vv) {
        float s = 0.0f;
#pragma unroll
        for (int nt = 0; nt < 16; ++nt) {
            float e = __expf(acc[nt][vv] - rmax[vv]);
            acc[nt][vv] = e;
            s += e;
        }
#pragma unroll
        for (int x = 1; x < 16; x <<= 1) s += __shfl_xor(s, x, 32);
        if (ln == 0) red[wave][vv + 8 * g] = s;
    }
    __syncthreads();
#pragma unroll
    for (int vv = 0; vv < 8; ++vv) {
        float s = red[0][vv + 8 * g];
#pragma unroll
        for (int w = 1; w < 4; ++w) s += red[w][vv + 8 * g];
        rinv[vv] = 1.0f / s;
    }

    // Write normalized attention (overwrites the bias scratch in place).
    float* arow = attn + (size_t)b * SS * SS;
#pragma unroll
    for (int nt = 0; nt < 16; ++nt) {
        const int cb = wave * 256 + nt * 16 + ln;
#pragma unroll
        for (int vv = 0; vv < 8; ++vv)
            arow[(size_t)(q0 + vv + 8 * g) * SS + cb] = acc[nt][vv] * rinv[vv];
    }
}

// ---------------------------------------------------------------------------
// Kernel 3: relative-value bias.  For fixed qpos:
//   out[b, qpos, d] = sum_k attn[b,qpos,k] * R_v[qpos,k,d]
// GEMM over batches: M=64(b), N=64(d), K=1024.  R_v read ONCE total.
// Grid: 1024 qpos, 128 threads = 4 waves; wave w owns batch tile w.
// Initializes out (kernel 4 accumulates on top).
// ---------------------------------------------------------------------------
__global__ __launch_bounds__(128) void k_bias_v(
    const float* __restrict__ attn, const float* __restrict__ Rv,
    float* __restrict__ out)
{
    const int qpos = blockIdx.x;
    const int tid  = threadIdx.x;
    const int wave = tid >> 5;
    const int lane = tid & 31;
    const int g    = lane >> 4;
    const int ln   = lane & 15;

    v8f acc[4];
#pragma unroll
    for (int nt = 0; nt < 4; ++nt) acc[nt] = {};

    for (int kb = 0; kb < SS; kb += 4) {
        const int kk = kb + 2 * g;
        v2f af = *(const v2f*)(attn + (size_t)(16 * wave + ln) * SS * SS
                                    + (size_t)qpos * SS + kk);
        const float* rvp = Rv + ((size_t)qpos * SS + kk) * DD + ln;
#pragma unroll
        for (int nt = 0; nt < 4; ++nt) {
            v2f bf;
            bf.x = rvp[nt * 16];
            bf.y = rvp[nt * 16 + DD];
            acc[nt] = wmma4(af, bf, acc[nt]);
        }
    }

#pragma unroll
    for (int nt = 0; nt < 4; ++nt)
#pragma unroll
        for (int vv = 0; vv < 8; ++vv) {
            const int b = 16 * wave + vv + 8 * g;
            out[((size_t)b * SS + qpos) * DD + nt * 16 + ln] = acc[nt][vv];
        }
}

// ---------------------------------------------------------------------------
// Kernel 4: out[b, q0:q0+16, :] += attn[b, q0:q0+16, :] @ V[b]
// Grid: (64 q-tiles, 64 batches), 128 threads = 4 waves; wave = n-tile.
// ---------------------------------------------------------------------------
__global__ __launch_bounds__(128) void k_attn_v(
    const float* __restrict__ attn, const float* __restrict__ vmat,
    float* __restrict__ out)
{
    const int q0   = blockIdx.x * 16;
    const int b    = blockIdx.y;
    const int tid  = threadIdx.x;
    const int nt   = tid >> 5;              // n-tile (16 of the 64 d cols)
    const int lane = tid & 31;
    const int g    = lane >> 4;
    const int ln   = lane & 15;

    v8f acc = {};
    for (int kb = 0; kb < SS; kb += 4) {
        const int kk = kb + 2 * g;
        v2f af = *(const v2f*)(attn + (size_t)b * SS * SS
                                    + (size_t)(q0 + ln) * SS + kk);
        const float* vp = vmat + ((size_t)b * SS + kk) * DD + nt * 16 + ln;
        v2f bf;
        bf.x = vp[0];
        bf.y = vp[DD];
        acc = wmma4(af, bf, acc);
    }

#pragma unroll
    for (int vv = 0; vv < 8; ++vv) {
        const int row = q0 + vv + 8 * g;
        const size_t idx = ((size_t)b * SS + row) * DD + nt * 16 + ln;
        out[idx] += acc[vv];
    }
}

// ---------------------------------------------------------------------------
extern "C" void kernel_launch(void* const* d_in, const int* in_sizes, int n_in,
                              void* d_out, int out_size, void* d_ws, size_t ws_size,
                              hipStream_t stream) {
    const float* q    = (const float*)d_in[0];
    const float* k    = (const float*)d_in[1];
    const float* vmat = (const float*)d_in[2];
    const float* Rk   = (const float*)d_in[3];
    const float* Rv   = (const float*)d_in[4];
    // d_in[5] = mask (implied by one_direction causal), d_in[6] = one_direction
    const int*   od   = (const int*)d_in[6];

    float* out  = (float*)d_out;                        // (B,S,D)
    float* attn = out + (size_t)BB * SS * DD;           // (B,S,S), also scores scratch

    // 1) bias scores into attn scratch (R_k streamed exactly once, async->LDS)
    k_bias_scores<<<dim3(8, SS), 256, 0, stream>>>(q, Rk, attn);
    // 2) qk + bias + mask + softmax -> final attn
    k_qk_softmax<<<dim3(SS / 16, BB), 128, 0, stream>>>(q, k, od, attn);
    // 3) out = attn @ R_v  (per-q GEMM over batches; R_v streamed once)
    k_bias_v<<<dim3(SS), 128, 0, stream>>>(attn, Rv, out);
    // 4) out += attn @ V
    k_attn_v<<<dim3(SS / 16, BB), 128, 0, stream>>>(attn, vmat, out);
}